// CombinedFocalLoss_64458869178583
// MI455X (gfx1250) — compile-verified
//
#include <hip/hip_runtime.h>
#include <math.h>

// ---------------------------------------------------------------------------
// Problem constants (from reference)
// ---------------------------------------------------------------------------
#define B_      32
#define H_      96
#define W_      96
#define HW_     (H_ * W_)          // 9216
#define C_      256
#define NHM     (B_ * HW_)         // 294912  (hm / gts elements)
#define NOFF    (B_ * 2 * HW_)     // 589824  (offset elements)
#define EPS_FOCAL  0.35f
#define NOISE_DIST 0.2f
// pos-loss coefficient: (1-eps) + eps*noise = 0.65 + 0.07 = 0.72
#define POS_COEF   0.72f

// workspace layout (floats)
//  [0] focal num_pos   [1] focal pos_sum   [2] focal neg_sum
//  [3] offset sse      [4] offset n_coefs
//  [5] gram pos-tile sum   [6] gram neg-tile sum
//  [7] cstency bce sum
//  [16..47]  norms[32]
//  [48..79]  (as int) argmax idx[32]
//  [80..80+32*256) feat[32][256]
#define WS_NORM 16
#define WS_IDX  48
#define WS_FEAT 80

typedef __attribute__((ext_vector_type(2))) float v2f;
typedef __attribute__((ext_vector_type(8))) float v8f;

// ---------------------------------------------------------------------------
// helpers
// ---------------------------------------------------------------------------
__device__ __forceinline__ float block_reduce_sum(float v, float* sbuf) {
    const int t = threadIdx.x;
    sbuf[t] = v;
    __syncthreads();
    for (int s = blockDim.x >> 1; s > 0; s >>= 1) {
        if (t < s) sbuf[t] += sbuf[t + s];
        __syncthreads();
    }
    float r = sbuf[0];
    __syncthreads();
    return r;
}

__device__ __forceinline__ float wave_reduce_sum(float v) {
    for (int off = 16; off > 0; off >>= 1) v += __shfl_xor(v, off, 32);
    return v;
}

// ---------------------------------------------------------------------------
// 0) zero the accumulator slots (harness poisons ws)
// ---------------------------------------------------------------------------
__global__ void init_kernel(float* ws) {
    if (threadIdx.x < 16) ws[threadIdx.x] = 0.0f;
}

// ---------------------------------------------------------------------------
// 1) focal loss partial sums
// ---------------------------------------------------------------------------
__global__ __launch_bounds__(256) void focal_kernel(const float* __restrict__ x,
                                                    const float* __restrict__ gt,
                                                    float* ws) {
    float np = 0.f, ps = 0.f, ns = 0.f;
    for (int i = blockIdx.x * blockDim.x + threadIdx.x; i < NHM;
         i += gridDim.x * blockDim.x) {
        float p = 1.0f / (1.0f + expf(-x[i]));
        p = fminf(fmaxf(p, 1e-4f), 1.0f - 1e-4f);
        float g = gt[i];
        if (g == 1.0f) {
            float omp = 1.0f - p;
            np += 1.0f;
            ps += POS_COEF * logf(p) * omp * omp;
        } else {                       // gt < 1 -> neg branch
            float w = 1.0f - g;
            w = w * w; w = w * w;      // (1-gt)^4
            ns += log1pf(-p) * p * p * w;
        }
    }
    __shared__ float sbuf[256];
    np = block_reduce_sum(np, sbuf);
    ps = block_reduce_sum(ps, sbuf);
    ns = block_reduce_sum(ns, sbuf);
    if (threadIdx.x == 0) {
        atomicAdd(&ws[0], np);
        atomicAdd(&ws[1], ps);
        atomicAdd(&ws[2], ns);
    }
}

// ---------------------------------------------------------------------------
// 2) per-row L2 norms of hm_outputs (32 rows x 9216)
// ---------------------------------------------------------------------------
__global__ __launch_bounds__(256) void norm_kernel(const float* __restrict__ hm,
                                                   float* ws) {
    const int b = blockIdx.x;
    const float* row = hm + (size_t)b * HW_;
    float s = 0.f;
    for (int i = threadIdx.x; i < HW_; i += blockDim.x) {
        float v = row[i];
        s = fmaf(v, v, s);
    }
    __shared__ float sbuf[256];
    s = block_reduce_sum(s, sbuf);
    if (threadIdx.x == 0) ws[WS_NORM + b] = fmaxf(sqrtf(s), 1e-6f);
}

// ---------------------------------------------------------------------------
// 3) gram tiles via V_WMMA_F32_16X16X4_F32
//    grid: 64 blocks x 32 threads (one wave each)
//      block = (tile, kchunk): tile 0 -> S[0:16,0:16], tile 1 -> S[0:16,16:32]
//      kchunk in [0,32): K range [kc*288, kc*288+288)
//    partial gram is normalized per-element (division is linear in the K-sum)
//    and atomically accumulated into ws[5]/ws[6].
// ---------------------------------------------------------------------------
__global__ __launch_bounds__(32) void gram_wmma_kernel(const float* __restrict__ hm,
                                                       float* ws) {
    const int tile = blockIdx.x & 1;
    const int kc   = blockIdx.x >> 1;          // 0..31
    const int k0   = kc * (HW_ / 32);          // chunks of 288
    const int k1   = k0 + (HW_ / 32);

    const int lane = threadIdx.x;              // 0..31, full wave, EXEC all-1s
    const int half = lane >> 4;                // 0 | 1
    const int l    = lane & 15;

    // A: rows 0..15 of hm;  B: rows tile*16 .. tile*16+15
    const float* Arow = hm + (size_t)l * HW_;
    const float* Brow = hm + (size_t)(tile * 16 + l) * HW_;

    v8f acc = {};
    for (int k = k0; k < k1; k += 4) {
        const int ka = k + half * 2;           // lanes 16-31 carry K=2,3
        v2f a = *(const v2f*)(Arow + ka);
        v2f b = *(const v2f*)(Brow + ka);
        acc = __builtin_amdgcn_wmma_f32_16x16x4_f32(
            /*neg_a=*/false, a, /*neg_b=*/false, b,
            /*c_mod=*/(short)0, acc, /*reuse_a=*/false, /*reuse_b=*/false);
    }

    // D layout: N = lane&15, M = r + 8*(lane>=16)
    const float invNn = 1.0f / ws[WS_NORM + tile * 16 + l];
    float s = 0.f;
#pragma unroll
    for (int r = 0; r < 8; ++r) {
        const int M = r + 8 * half;
        s += acc[r] * invNn / ws[WS_NORM + M];
    }
    s = wave_reduce_sum(s);
    if (lane == 0) atomicAdd(&ws[5 + tile], s);
}

// ---------------------------------------------------------------------------
// 4) masked offset MSE partial sums
// ---------------------------------------------------------------------------
__global__ __launch_bounds__(256) void offset_kernel(const float* __restrict__ p,
                                                     const float* __restrict__ g,
                                                     float* ws) {
    float sse = 0.f, n = 0.f;
    for (int i = blockIdx.x * blockDim.x + threadIdx.x; i < NOFF;
         i += gridDim.x * blockDim.x) {
        float gv = g[i];
        if (gv > 0.0f) {
            float d = p[i] - gv;
            sse = fmaf(d, d, sse);
            n += 1.0f;
        }
    }
    __shared__ float sbuf[256];
    sse = block_reduce_sum(sse, sbuf);
    n   = block_reduce_sum(n, sbuf);
    if (threadIdx.x == 0) {
        atomicAdd(&ws[3], sse);
        atomicAdd(&ws[4], n);
    }
}

// ---------------------------------------------------------------------------
// 5) per-batch argmax of cstency_gts (first-max semantics like jnp.argmax)
// ---------------------------------------------------------------------------
__global__ __launch_bounds__(256) void argmax_kernel(const float* __restrict__ gts,
                                                     float* ws) {
    const int b = blockIdx.x;
    const float* g = gts + (size_t)b * HW_;
    float best = -INFINITY;
    int bi = 0;
    for (int i = threadIdx.x; i < HW_; i += blockDim.x) {
        float v = g[i];
        if (v > best) { best = v; bi = i; }   // i increasing -> first occurrence
    }
    __shared__ float sv[256];
    __shared__ int   si[256];
    const int t = threadIdx.x;
    sv[t] = best; si[t] = bi;
    __syncthreads();
    for (int s = 128; s > 0; s >>= 1) {
        if (t < s) {
            float ov = sv[t + s]; int oi = si[t + s];
            if (ov > sv[t] || (ov == sv[t] && oi < si[t])) { sv[t] = ov; si[t] = oi; }
        }
        __syncthreads();
    }
    if (t == 0) ((int*)(ws + WS_IDX))[b] = si[0];
}

// ---------------------------------------------------------------------------
// 6) gather feat[b][c] = preds[b, c, idx[b]]
// ---------------------------------------------------------------------------
__global__ __launch_bounds__(256) void gather_feat_kernel(const float* __restrict__ preds,
                                                          float* ws) {
    const int b = blockIdx.x;
    const int c = threadIdx.x;
    const int idx = ((const int*)(ws + WS_IDX))[b];
    ws[WS_FEAT + b * C_ + c] =
        preds[((size_t)b * C_ + c) * HW_ + idx];
}

// ---------------------------------------------------------------------------
// 7) cstency: m[b,n] = feat[b,:]·preds[b,:,n]/16 ; bce(sigmoid(m), gts)
//    This streams the 302 MB tensor once, fully coalesced -> bandwidth bound.
//    block = (batch, 256-wide n-chunk); feat vector staged in LDS.
// ---------------------------------------------------------------------------
__global__ __launch_bounds__(256) void cstency_kernel(const float* __restrict__ preds,
                                                      const float* __restrict__ gts,
                                                      float* ws) {
    __shared__ float fs[C_];
    const int b     = blockIdx.x / (HW_ / 256);   // 36 chunks per batch
    const int chunk = blockIdx.x % (HW_ / 256);
    fs[threadIdx.x] = ws[WS_FEAT + b * C_ + threadIdx.x];
    __syncthreads();

    const int n = chunk * 256 + threadIdx.x;
    const float* p = preds + (size_t)b * C_ * HW_ + n;
    float m = 0.f;
#pragma unroll 8
    for (int c = 0; c < C_; ++c)
        m = fmaf(fs[c], p[(size_t)c * HW_], m);
    m *= (1.0f / 16.0f);                          // / sqrt(C)

    float enc = 1.0f / (1.0f + expf(-m));
    enc = fminf(fmaxf(enc, 1e-7f), 1.0f - 1e-7f);
    const float g = gts[(size_t)b * HW_ + n];
    float term = -(g * logf(enc) + (1.0f - g) * log1pf(-enc));

    __shared__ float sbuf[256];
    term = block_reduce_sum(term, sbuf);
    if (threadIdx.x == 0) atomicAdd(&ws[7], term);
}

// ---------------------------------------------------------------------------
// 8) final assembly (one wave): cls BCE + combine all partials
// ---------------------------------------------------------------------------
__global__ __launch_bounds__(32) void final_kernel(const float* __restrict__ cls_p,
                                                   const float* __restrict__ cls_g,
                                                   const float* __restrict__ ws,
                                                   float* out) {
    const int t = threadIdx.x;                    // 32 threads == B
    float p = fminf(fmaxf(cls_p[t], 1e-7f), 1.0f - 1e-7f);
    float g = cls_g[t];
    float term = -(g * logf(p) + (1.0f - g) * log1pf(-p));
    term = wave_reduce_sum(term);
    if (t == 0) {
        const float num_pos = ws[0], ps = ws[1], ns = ws[2];
        // loss_hm
        out[0] = (num_pos == 0.0f) ? -ns : -(ps + ns) / fmaxf(num_pos, 1.0f);
        // loss_cls
        out[1] = term / (float)B_;
        // loss_dst: (pos - neg)/(h*h)*alpha = 0.5*(Sneg - Spos)/256 * 0.01
        out[2] = 0.5f * (ws[6] - ws[5]) / 256.0f * 0.01f;
        // loss_off
        out[3] = 0.5f * ws[3] / (ws[4] + 1e-6f);
        // loss_cst
        out[4] = ws[7] / (float)NHM;
    }
}

// ---------------------------------------------------------------------------
// launch
// ---------------------------------------------------------------------------
extern "C" void kernel_launch(void* const* d_in, const int* in_sizes, int n_in,
                              void* d_out, int out_size, void* d_ws, size_t ws_size,
                              hipStream_t stream) {
    (void)in_sizes; (void)n_in; (void)out_size; (void)ws_size;
    const float* hm_outputs    = (const float*)d_in[0];
    const float* hm_targets    = (const float*)d_in[1];
    const float* cls_preds     = (const float*)d_in[2];
    const float* cls_gts       = (const float*)d_in[3];
    const float* offset_preds  = (const float*)d_in[4];
    const float* offset_gts    = (const float*)d_in[5];
    const float* cstency_preds = (const float*)d_in[6];
    const float* cstency_gts   = (const float*)d_in[7];
    float* out = (float*)d_out;
    float* ws  = (float*)d_ws;

    init_kernel<<<1, 32, 0, stream>>>(ws);
    focal_kernel<<<288, 256, 0, stream>>>(hm_outputs, hm_targets, ws);
    norm_kernel<<<B_, 256, 0, stream>>>(hm_outputs, ws);
    gram_wmma_kernel<<<64, 32, 0, stream>>>(hm_outputs, ws);
    offset_kernel<<<576, 256, 0, stream>>>(offset_preds, offset_gts, ws);
    argmax_kernel<<<B_, 256, 0, stream>>>(cstency_gts, ws);
    gather_feat_kernel<<<B_, C_, 0, stream>>>(cstency_preds, ws);
    cstency_kernel<<<B_ * (HW_ / 256), 256, 0, stream>>>(cstency_preds, cstency_gts, ws);
    final_kernel<<<1, 32, 0, stream>>>(cls_preds, cls_gts, ws, out);
}